// Embeddings_74156905333343
// MI455X (gfx1250) — compile-verified
//
#include <hip/hip_runtime.h>
#include <math.h>

// Embedding fuse for MI455X (gfx1250, wave32).
// Memory-bound: ~64MB out + ~70MB gathered reads @ 23.3 TB/s -> ~6us floor.
// No matmul => no WMMA. CDNA5-specific path used instead: async global->LDS
// copies (ASYNCcnt) to double-buffer the dependent token-row gathers, and
// non-temporal b128 stores so the 64MB output doesn't evict the 131MB token
// table from the 192MB L2.

typedef float v4f __attribute__((ext_vector_type(4)));

// ---- CDNA5 async global->LDS copy (16B per lane), inline asm for
// ---- portability across ROCm7.2 / amdgpu-toolchain (bypasses builtins).
__device__ __forceinline__ void async_copy16(unsigned lds_byte_addr, const void* gaddr) {
  asm volatile("global_load_async_to_lds_b128 %0, %1, off"
               :: "v"(lds_byte_addr), "v"(gaddr)
               : "memory");
}

__device__ __forceinline__ void wait_async_le(int n) {
#if __has_builtin(__builtin_amdgcn_s_wait_asynccnt)
  if (n <= 0) __builtin_amdgcn_s_wait_asynccnt(0);
  else        __builtin_amdgcn_s_wait_asynccnt(1);
#else
  if (n <= 0) asm volatile("s_wait_asynccnt 0x0" ::: "memory");
  else        asm volatile("s_wait_asynccnt 0x1" ::: "memory");
#endif
}

// -------------------------------------------------------------------------
// Kernel 1: seg[b,s] = cummax_s(x[b,s]==sep). One wave32 per batch row,
// prefix-OR via __ballot. Trivial cost (16K elements).
// -------------------------------------------------------------------------
__global__ void seg_cummax_kernel(const int* __restrict__ x,
                                  const int* __restrict__ sep_ptr,
                                  unsigned char* __restrict__ seg,
                                  int S) {
  const int b    = blockIdx.x;
  const int lane = threadIdx.x;          // 0..31
  const int sep  = sep_ptr[0];
  const int* row = x + (size_t)b * S;
  unsigned char* srow = seg + (size_t)b * S;

  unsigned carry = 0u;
  for (int s0 = 0; s0 < S; s0 += 32) {
    const int s = s0 + lane;
    const int flag = (s < S) ? (row[s] == sep) : 0;
    const unsigned m = (unsigned)__ballot(flag);      // wave32: low 32 bits
    const unsigned incl = m & ((2u << lane) - 1u);    // inclusive prefix mask
    if (s < S) srow[s] = (unsigned char)((carry | incl) ? 1u : 0u);
    carry |= m;                                       // uniform across wave
  }
}

// -------------------------------------------------------------------------
// Kernel 2: one 256-thread block per sequence position s.
//   LDS layout (D=1024 floats per row): [pos | tokbuf0 | tokbuf1] = 12KB.
//   pos row staged once; token rows for b=0..B-1 double-buffered through the
//   async copy engine, overlapping the next gather with the current compute.
//   Each thread owns bytes [tid*16, tid*16+16) of every row -> no barriers,
//   only per-wave s_wait_asynccnt.
// -------------------------------------------------------------------------
__global__ void embed_fuse_kernel(const int* __restrict__ x,
                                  const float* __restrict__ tok_table,
                                  const float* __restrict__ pos_table,
                                  const float* __restrict__ seg_table,
                                  const unsigned char* __restrict__ seg,
                                  float* __restrict__ out,
                                  int B, int S, int D, float scale) {
  __shared__ __align__(16) float smem[3 * 1024];   // D == 1024 for this problem

  const int s   = blockIdx.x;            // 0..S-1
  const int tid = threadIdx.x;           // 0..D/4-1 (256)

  // Low 32 bits of a generic LDS pointer == workgroup-relative LDS byte addr.
  const unsigned lbase = (unsigned)(size_t)(void*)&smem[0];
  const unsigned loff  = (unsigned)tid * 16u;
  const unsigned rowB  = (unsigned)D * 4u;          // bytes per staged row

  // Stage pos row (buffer 0 region) and token row for b=0 (tokbuf0).
  async_copy16(lbase + loff, pos_table + (size_t)s * D + tid * 4);
  const int token0 = x[s];
  async_copy16(lbase + rowB + loff, tok_table + (size_t)token0 * D + tid * 4);

  for (int b = 0; b < B; ++b) {
    if (b + 1 < B) {
      // Prefetch next token row into the other buffer while we compute.
      const int tok_n = x[(size_t)(b + 1) * S + s];
      async_copy16(lbase + rowB * (unsigned)(1 + ((b + 1) & 1)) + loff,
                   tok_table + (size_t)tok_n * D + tid * 4);
      wait_async_le(1);   // pos + tok[b] complete (in-order); prefetch in flight
    } else {
      wait_async_le(0);
    }

    const v4f p = *(const v4f*)&smem[tid * 4];
    const v4f t = *(const v4f*)&smem[(1 + (b & 1)) * D + tid * 4];

    const int sg = seg[(size_t)b * S + s];
    const v4f g = *(const v4f*)(seg_table + (size_t)sg * D + tid * 4);

    v4f r;
    r.x = (t.x + p.x + g.x) * scale;
    r.y = (t.y + p.y + g.y) * scale;
    r.z = (t.z + p.z + g.z) * scale;
    r.w = (t.w + p.w + g.w) * scale;

    // Output is write-once: non-temporal so it doesn't thrash L2.
    v4f* dst = (v4f*)(out + ((size_t)b * S + s) * D + tid * 4);
    __builtin_nontemporal_store(r, dst);
  }
}

// -------------------------------------------------------------------------
extern "C" void kernel_launch(void* const* d_in, const int* in_sizes, int n_in,
                              void* d_out, int out_size, void* d_ws, size_t ws_size,
                              hipStream_t stream) {
  const int*   x    = (const int*)d_in[0];     // [B,S] int32
  const int*   sep  = (const int*)d_in[1];     // scalar
  const float* tokt = (const float*)d_in[2];   // [V,D]
  const float* post = (const float*)d_in[3];   // [S,D]
  const float* segt = (const float*)d_in[4];   // [2,D]
  float*       out  = (float*)d_out;           // [B,S,D]

  const int D = in_sizes[4] / 2;               // 1024
  const int S = in_sizes[3] / D;               // 2048
  const int B = in_sizes[0] / S;               // 8
  const float scale = sqrtf((float)D);         // 32.0

  unsigned char* segbuf = (unsigned char*)d_ws;   // B*S bytes

  seg_cummax_kernel<<<B, 32, 0, stream>>>(x, sep, segbuf, S);
  embed_fuse_kernel<<<S, D / 4, 0, stream>>>(x, tokt, post, segt, segbuf, out,
                                             B, S, D, scale);
}